// LightGCN_35708358099477
// MI455X (gfx1250) — compile-verified
//
#include <hip/hip_runtime.h>
#include <hip/hip_bf16.h>

// ---------------------------------------------------------------------------
// LightGCN 2-layer smoothing, CSR-based (atomic-free SpMM passes).
//
// Pipeline (all kernels serialized on `stream`):
//   1) zero degree counters
//   2) degree histogram (u32 atomics, 2 per edge)
//   3) dinv[n] = rsqrt(deg)  (0 if isolated)
//   4) block sums of deg                      \
//   5) scan of block sums (1 block, LDS)       > exclusive prefix -> rowptr
//   6) per-block scan + offset -> rowptr,cursor/
//   7) fill CSR (col=src, val=dinv[src]*dinv[dst]) via atomic cursors
//   8) smooth pass 1: x1[n] = sum_j w_j * x0[src_j]      (wave per row)
//   9) smooth pass 2 fused with combine:
//        out[n] = (2*x0[n] + 2*x1[n] + sum_j w_j * x1[src_j]) / 3
//
// The gather in (8)/(9) is staged through LDS with CDNA5 async-tensor ops:
// GLOBAL_LOAD_ASYNC_TO_LDS_B128 stages two full 256B rows per instruction
// (lanes 0-15 -> row k, lanes 16-31 -> row k+1) on the ASYNCcnt path, with
// ping-pong double buffering: chunk i+1 is issued before chunk i is consumed
// and s_wait_asynccnt <= 8 (one full chunk of ops, in-order completion)
// guarantees chunk i has landed.
// ---------------------------------------------------------------------------

#define EMB_D        64
#define ROW_BYTES    (EMB_D * 4)  // 256B per embedding row
#define SCAN_BS      256          // elements per scan block
#define SCAN2_BS     1024         // max #scan blocks handled (N <= 262144)
#define WAVE_SZ      32           // gfx1250 is wave32
#define WAVES_PER_BLK 8           // 256-thread blocks
#define SLOTS        16           // staged rows per wave per chunk buffer

static_assert(SLOTS == 16, "s_wait_asynccnt immediate below assumes 8 ops/chunk");

__global__ void k_zero_u32(unsigned* __restrict__ p, int n) {
    int i = blockIdx.x * blockDim.x + threadIdx.x;
    if (i < n) p[i] = 0u;
}

__global__ void k_degree(const int* __restrict__ u_idx,
                         const int* __restrict__ i_idx,
                         unsigned* __restrict__ deg, int E, int NU) {
    int e = blockIdx.x * blockDim.x + threadIdx.x;
    if (e < E) {
        atomicAdd(&deg[u_idx[e]], 1u);
        atomicAdd(&deg[NU + i_idx[e]], 1u);
    }
}

__global__ void k_dinv(const unsigned* __restrict__ deg,
                       float* __restrict__ dinv, int N) {
    int n = blockIdx.x * blockDim.x + threadIdx.x;
    if (n < N) {
        unsigned d = deg[n];
        dinv[n] = d ? rsqrtf((float)d) : 0.0f;
    }
}

// per-block reduction of deg -> bsum[block]
__global__ void k_scan_block_sums(const unsigned* __restrict__ deg,
                                  unsigned* __restrict__ bsum, int N) {
    __shared__ unsigned s[SCAN_BS];
    int i = blockIdx.x * SCAN_BS + threadIdx.x;
    s[threadIdx.x] = (i < N) ? deg[i] : 0u;
    __syncthreads();
    for (int off = SCAN_BS / 2; off > 0; off >>= 1) {
        if (threadIdx.x < off) s[threadIdx.x] += s[threadIdx.x + off];
        __syncthreads();
    }
    if (threadIdx.x == 0) bsum[blockIdx.x] = s[0];
}

// single-block exclusive scan of block sums (nb <= SCAN2_BS)
__global__ void k_scan_bsums(unsigned* __restrict__ bsum, int nb) {
    __shared__ unsigned s[SCAN2_BS];
    int t = threadIdx.x;
    unsigned v = (t < nb) ? bsum[t] : 0u;
    s[t] = v;
    __syncthreads();
    for (int off = 1; off < SCAN2_BS; off <<= 1) {
        unsigned add = (t >= off) ? s[t - off] : 0u;
        __syncthreads();
        s[t] += add;
        __syncthreads();
    }
    if (t < nb) bsum[t] = s[t] - v;   // exclusive
}

// per-block exclusive scan of deg + block offset -> rowptr, cursor
__global__ void k_rowptr(const unsigned* __restrict__ deg,
                         const unsigned* __restrict__ bsum,
                         unsigned* __restrict__ rowptr,
                         unsigned* __restrict__ cursor,
                         int N, unsigned DE) {
    __shared__ unsigned s[SCAN_BS];
    int i = blockIdx.x * SCAN_BS + threadIdx.x;
    unsigned v = (i < N) ? deg[i] : 0u;
    s[threadIdx.x] = v;
    __syncthreads();
    for (int off = 1; off < SCAN_BS; off <<= 1) {
        unsigned add = (threadIdx.x >= off) ? s[threadIdx.x - off] : 0u;
        __syncthreads();
        s[threadIdx.x] += add;
        __syncthreads();
    }
    if (i < N) {
        unsigned ex = bsum[blockIdx.x] + s[threadIdx.x] - v;  // exclusive
        rowptr[i] = ex;
        cursor[i] = ex;
    }
    if (i == 0) rowptr[N] = DE;
}

// scatter (src, w) into CSR slots of dst via atomic cursors
__global__ void k_fill(const int* __restrict__ u_idx,
                       const int* __restrict__ i_idx,
                       const float* __restrict__ dinv,
                       unsigned* __restrict__ cursor,
                       unsigned* __restrict__ col,
                       float* __restrict__ val,
                       int E, int NU) {
    int de = blockIdx.x * blockDim.x + threadIdx.x;
    if (de >= 2 * E) return;
    int src, dst;
    if (de < E) { src = u_idx[de];          dst = NU + i_idx[de]; }
    else        { int e = de - E; src = NU + i_idx[e]; dst = u_idx[e]; }
    float w = dinv[src] * dinv[dst];
    unsigned pos = atomicAdd(&cursor[dst], 1u);
    col[pos] = (unsigned)src;
    val[pos] = w;
}

// Wave-per-row weighted row-sum with double-buffered async-to-LDS gathers.
// Source row r < NU comes from xa + r*64, otherwise xb + (r-NU)*64.
// If final_combine: fold (2*x0 + 2*x1 + acc)/3.
__global__ void __launch_bounds__(WAVE_SZ * WAVES_PER_BLK)
k_smooth(const unsigned* __restrict__ rowptr,
         const unsigned* __restrict__ col,
         const float* __restrict__ val,
         const float* __restrict__ xa,     // gather source, rows [0, NU)
         const float* __restrict__ xb,     // gather source, rows [NU, N)
         float* __restrict__ y,            // output rows
         int N, int NU, int final_combine,
         const float* __restrict__ u_emb,  // only for final combine
         const float* __restrict__ i_emb,
         const float* __restrict__ x1buf) {
    __shared__ float stage[WAVES_PER_BLK][2][SLOTS][EMB_D];   // 64KB / block
    const int wv   = threadIdx.x >> 5;
    const int lane = threadIdx.x & (WAVE_SZ - 1);
    const int n    = (int)(blockIdx.x * WAVES_PER_BLK + wv);
    if (n >= N) return;

    const unsigned beg = rowptr[n];
    const unsigned end = rowptr[n + 1];
    const unsigned len = end - beg;
    const int c = lane * 2;                 // this lane owns columns c, c+1
    float2 acc; acc.x = 0.0f; acc.y = 0.0f;

    // LDS byte offset of this wave's staging buffers (low 32 bits of the
    // flat address are the LDS offset on gfx1250).
    const unsigned lds_base = (unsigned)(uintptr_t)(&stage[wv][0][0][0]);
    const int half = lane >> 4;             // 0: rows 2t, 1: rows 2t+1
    const int q    = lane & 15;             // 16B slice (1/16 of a row) owner

    // Issue one chunk of up to SLOTS rows into buffer `buf`.
    // Each b128 async op stages TWO rows (16 lanes x 16B each).
    // Returns this lane's staged edge weight (for shfl at consume time).
    auto issue_chunk = [&](unsigned ci, int buf) -> float {
        unsigned base = beg + ci * SLOTS;
        int nn = (int)(end - base);
        if (nn > SLOTS) nn = SLOTS;
        unsigned jj = base + (unsigned)lane;
        unsigned cj = (lane < nn) ? col[jj] : 0u;
        float    wj = (lane < nn) ? val[jj] : 0.0f;
        if (base + SLOTS < end)                       // next chunk metadata
            __builtin_prefetch(&col[base + SLOTS], 0, 1);
        int nops = (nn + 1) >> 1;
        for (int t = 0; t < nops; ++t) {
            int k = 2 * t + half;                     // row index in chunk
            int kd = (k >= nn) ? (nn - 1) : k;        // clamp data (odd tail)
            unsigned src = (unsigned)__shfl((int)cj, kd, WAVE_SZ);
            const float* xrow = (src < (unsigned)NU)
                              ? (xa + (size_t)src * EMB_D)
                              : (xb + (size_t)(src - (unsigned)NU) * EMB_D);
            const void* g = (const void*)((const char*)xrow + q * 16);
            unsigned l = lds_base + (unsigned)buf * (SLOTS * ROW_BYTES)
                       + (unsigned)k * ROW_BYTES + (unsigned)(q * 16);
            asm volatile("global_load_async_to_lds_b128 %0, %1, off"
                         :: "v"(l), "v"(g) : "memory");
        }
        return wj;
    };

    auto consume_chunk = [&](unsigned ci, int buf, float wj) {
        unsigned base = beg + ci * SLOTS;
        int nn = (int)(end - base);
        if (nn > SLOTS) nn = SLOTS;
        for (int k = 0; k < nn; ++k) {
            float w = __shfl(wj, k, WAVE_SZ);
            float2 xv = *(const float2*)(&stage[wv][buf][k][c]);
            acc.x = fmaf(w, xv.x, acc.x);
            acc.y = fmaf(w, xv.y, acc.y);
        }
    };

    if (len > 0) {
        const unsigned nchunks = (len + SLOTS - 1) / SLOTS;
        float wcur = issue_chunk(0, 0);
        for (unsigned i = 0; i < nchunks; ++i) {
            float wnxt = 0.0f;
            if (i + 1 < nchunks) {
                wnxt = issue_chunk(i + 1, (int)((i + 1) & 1));
                if (len - (i + 1) * SLOTS >= SLOTS) {
                    // next chunk is full (8 b128 ops): <=8 outstanding means
                    // chunk i fully landed (in-order async completion).
                    asm volatile("s_wait_asynccnt 0x8" ::: "memory");
                } else {
                    asm volatile("s_wait_asynccnt 0x0" ::: "memory");
                }
            } else {
                asm volatile("s_wait_asynccnt 0x0" ::: "memory");
            }
            consume_chunk(i, (int)(i & 1), wcur);
            wcur = wnxt;
        }
    }

    if (final_combine) {
        const float* x0row = (n < NU) ? (u_emb + (size_t)n * EMB_D)
                                      : (i_emb + (size_t)(n - NU) * EMB_D);
        float2 x0v = *(const float2*)(x0row + c);
        float2 x1v = *(const float2*)(x1buf + (size_t)n * EMB_D + c);
        const float k3 = 1.0f / 3.0f;
        acc.x = (2.0f * x0v.x + 2.0f * x1v.x + acc.x) * k3;
        acc.y = (2.0f * x0v.y + 2.0f * x1v.y + acc.y) * k3;
    }
    *(float2*)(y + (size_t)n * EMB_D + c) = acc;
}

// ---------------------------------------------------------------------------

static inline size_t align_up(size_t v, size_t a) { return (v + a - 1) & ~(a - 1); }

extern "C" void kernel_launch(void* const* d_in, const int* in_sizes, int n_in,
                              void* d_out, int out_size, void* d_ws, size_t ws_size,
                              hipStream_t stream) {
    const float* u_emb = (const float*)d_in[0];
    const float* i_emb = (const float*)d_in[1];
    const int*   u_idx = (const int*)d_in[2];
    const int*   i_idx = (const int*)d_in[3];
    float*       out   = (float*)d_out;

    const int NU = in_sizes[0] / EMB_D;
    const int NI = in_sizes[1] / EMB_D;
    const int E  = in_sizes[2];
    const int N  = NU + NI;
    const int DE = 2 * E;                  // directed edge count

    // ---- carve workspace -------------------------------------------------
    char* p = (char*)d_ws;
    size_t off = 0;
    auto carve = [&](size_t bytes) {
        void* r = p + off;
        off = align_up(off + bytes, 256);
        return r;
    };
    unsigned* deg    = (unsigned*)carve((size_t)N * 4);
    float*    dinv   = (float*)   carve((size_t)N * 4);
    unsigned* rowptr = (unsigned*)carve((size_t)(N + 1) * 4);
    unsigned* cursor = (unsigned*)carve((size_t)N * 4);
    unsigned* bsum   = (unsigned*)carve((size_t)SCAN2_BS * 4);
    unsigned* col    = (unsigned*)carve((size_t)DE * 4);
    float*    val    = (float*)   carve((size_t)DE * 4);
    float*    x1     = (float*)   carve((size_t)N * EMB_D * 4);
    (void)ws_size;

    const int BLK = 256;
    const int nb  = (N + SCAN_BS - 1) / SCAN_BS;   // 782 for N=200000

    // 1) zero degrees
    k_zero_u32<<<(N + BLK - 1) / BLK, BLK, 0, stream>>>(deg, N);
    // 2) degree histogram
    k_degree<<<(E + BLK - 1) / BLK, BLK, 0, stream>>>(u_idx, i_idx, deg, E, NU);
    // 3) dinv
    k_dinv<<<(N + BLK - 1) / BLK, BLK, 0, stream>>>(deg, dinv, N);
    // 4-6) exclusive scan deg -> rowptr (and cursor copy)
    k_scan_block_sums<<<nb, SCAN_BS, 0, stream>>>(deg, bsum, N);
    k_scan_bsums<<<1, SCAN2_BS, 0, stream>>>(bsum, nb);
    k_rowptr<<<nb, SCAN_BS, 0, stream>>>(deg, bsum, rowptr, cursor, N, (unsigned)DE);
    // 7) fill CSR
    k_fill<<<(DE + BLK - 1) / BLK, BLK, 0, stream>>>(u_idx, i_idx, dinv, cursor,
                                                     col, val, E, NU);
    // 8) pass 1: x1 = A_hat * x0   (x0 = concat(u_emb, i_emb))
    const int smooth_blocks = (N + WAVES_PER_BLK - 1) / WAVES_PER_BLK;
    k_smooth<<<smooth_blocks, WAVE_SZ * WAVES_PER_BLK, 0, stream>>>(
        rowptr, col, val, u_emb, i_emb, x1, N, NU, 0, nullptr, nullptr, nullptr);
    // 9) pass 2 fused with combine: out = (2*x0 + 2*x1 + A_hat*x1) / 3
    k_smooth<<<smooth_blocks, WAVE_SZ * WAVES_PER_BLK, 0, stream>>>(
        rowptr, col, val, x1, x1 + (size_t)NU * EMB_D, out, N, NU, 1,
        u_emb, i_emb, x1);
}